// EdgeNetwork_83030307766410
// MI455X (gfx1250) — compile-verified
//
#include <hip/hip_runtime.h>
#include <hip/hip_bf16.h>

// CDNA5 / gfx1250 edge-MLP. Wave32, WMMA f32 16x16x4, LDS-staged tiles.

typedef __attribute__((ext_vector_type(2))) float v2f;
typedef __attribute__((ext_vector_type(8))) float v8f;

#define WAVES_PER_BLOCK 8
#define TILES_PER_WAVE 8
#define TILE_M 16

__device__ __forceinline__ float fast_tanhf(float x) {
#if __has_builtin(__builtin_amdgcn_tanhf)
  return __builtin_amdgcn_tanhf(x);     // gfx1250 v_tanh_f32 (TRANS unit)
#elif __has_builtin(__builtin_amdgcn_tanh_f32)
  return __builtin_amdgcn_tanh_f32(x);
#else
  return tanhf(x);                      // ocml fallback
#endif
}

__device__ __forceinline__ v8f wmma_f32_k4(v2f a, v2f b, v8f c) {
  // D = A(16x4,f32) * B(4x16,f32) + C(16x16,f32)
  return __builtin_amdgcn_wmma_f32_16x16x4_f32(
      /*neg_a=*/false, a, /*neg_b=*/false, b,
      /*c_mod=*/(short)0, c, /*reuse_a=*/false, /*reuse_b=*/false);
}

__global__ __launch_bounds__(WAVES_PER_BLOCK * 32)
void edge_mlp_wmma(const float* __restrict__ feats,   // [N,16]
                   const int*   __restrict__ eidx,    // [2,E]
                   const float* __restrict__ W1, const float* __restrict__ b1,  // [32,8],[8]
                   const float* __restrict__ W2, const float* __restrict__ b2,  // [8,8],[8]
                   const float* __restrict__ W3, const float* __restrict__ b3,  // [8,8],[8]
                   const float* __restrict__ W4, const float* __restrict__ b4,  // [8,1],[1]
                   float* __restrict__ out, int nedges, int ntiles) {
  __shared__ float lds[WAVES_PER_BLOCK][TILE_M * 32];   // 2KB per wave
  const int lane = threadIdx.x & 31;
  const int wave = threadIdx.x >> 5;
  float* sm = lds[wave];

  const int n    = lane & 15;   // B/C column (and A row) owned by this lane
  const int half = lane >> 4;   // lane half selects K sub-pair / M offset
  const int kh   = half * 2;

  // ---- Pre-fragment weights into registers (B layout: lane=n, k = 4c+kh+{0,1}).
  // Columns 8..15 padded with zeros -> those outputs are exactly 0 through tanh.
  v2f w1f[8];
#pragma unroll
  for (int c = 0; c < 8; ++c) {
    const int k = 4 * c + kh;
    w1f[c].x = (n < 8) ? W1[(k + 0) * 8 + n] : 0.0f;
    w1f[c].y = (n < 8) ? W1[(k + 1) * 8 + n] : 0.0f;
  }
  v2f w2f[2], w3f[2];
#pragma unroll
  for (int c = 0; c < 2; ++c) {
    const int k = 4 * c + kh;
    w2f[c].x = (n < 8) ? W2[(k + 0) * 8 + n] : 0.0f;
    w2f[c].y = (n < 8) ? W2[(k + 1) * 8 + n] : 0.0f;
    w3f[c].x = (n < 8) ? W3[(k + 0) * 8 + n] : 0.0f;
    w3f[c].y = (n < 8) ? W3[(k + 1) * 8 + n] : 0.0f;
  }
  const float bias1 = (n < 8) ? b1[n] : 0.0f;
  const float bias2 = (n < 8) ? b2[n] : 0.0f;
  const float bias3 = (n < 8) ? b3[n] : 0.0f;
  const float w4n   = (n < 8) ? W4[n] : 0.0f;
  const float b4v   = b4[0];

  const int t0 = (blockIdx.x * WAVES_PER_BLOCK + wave) * TILES_PER_WAVE;
  for (int ti = 0; ti < TILES_PER_WAVE; ++ti) {
    const int t = t0 + ti;
    if (t >= ntiles) break;                 // uniform per wave
    const int e0 = t * TILE_M;

    // ---- Gather: lanes 0-15 -> src row of edge e0+n, lanes 16-31 -> dst row.
    // One contiguous 64B row per lane; node table is L2-resident (6.4MB).
    {
      int e = e0 + n;
      if (e >= nedges) e = nedges - 1;      // tail clamp (stores guarded below)
      const int node = half ? eidx[nedges + e] : eidx[e];
      const float4* src = (const float4*)(feats + (long long)node * 16);
      const float4 r0 = src[0], r1 = src[1], r2 = src[2], r3 = src[3];
      float4* dst = (float4*)(sm + n * 32 + half * 16);
      dst[0] = r0; dst[1] = r1; dst[2] = r2; dst[3] = r3;
    }
    __builtin_amdgcn_wave_barrier();        // single-wave LDS is HW in-order

    // ---- Layer 1: [16x32] @ [32x16pad] via 8 chained f32 WMMAs (K=4 each)
    v8f acc1 = {};
#pragma unroll
    for (int c = 0; c < 8; ++c) {
      const int k = 4 * c + kh;
      v2f a; a.x = sm[n * 32 + k]; a.y = sm[n * 32 + k + 1];
      acc1 = wmma_f32_k4(a, w1f[c], acc1);
    }
    __builtin_amdgcn_wave_barrier();

    // bias + hw tanh, transpose C (N-per-lane) -> A layout (M-per-lane) via LDS
    #pragma unroll
    for (int v = 0; v < 8; ++v) {
      const float h = fast_tanhf(acc1[v] + bias1);
      if (n < 8) sm[(v + 8 * half) * 8 + n] = h;   // h1: sm[0..127]
    }
    __builtin_amdgcn_wave_barrier();

    // ---- Layer 2: [16x8] @ [8x16pad], 2 WMMAs
    v8f acc2 = {};
#pragma unroll
    for (int c = 0; c < 2; ++c) {
      const int k = 4 * c + kh;
      v2f a; a.x = sm[n * 8 + k]; a.y = sm[n * 8 + k + 1];
      acc2 = wmma_f32_k4(a, w2f[c], acc2);
    }
    __builtin_amdgcn_wave_barrier();
    #pragma unroll
    for (int v = 0; v < 8; ++v) {
      const float h = fast_tanhf(acc2[v] + bias2);
      if (n < 8) sm[128 + (v + 8 * half) * 8 + n] = h;   // h2: sm[128..255]
    }
    __builtin_amdgcn_wave_barrier();

    // ---- Layer 3: [16x8] @ [8x16pad], 2 WMMAs
    v8f acc3 = {};
#pragma unroll
    for (int c = 0; c < 2; ++c) {
      const int k = 4 * c + kh;
      v2f a; a.x = sm[128 + n * 8 + k]; a.y = sm[128 + n * 8 + k + 1];
      acc3 = wmma_f32_k4(a, w3f[c], acc3);
    }

    // ---- Layer 4: out[m] = sum_n h3[m][n]*W4[n] + b4 (cross-lane reduce)
    #pragma unroll
    for (int v = 0; v < 8; ++v) {
      float p = fast_tanhf(acc3[v] + bias3) * w4n;   // pad lanes contribute 0
      p += __shfl_xor(p, 1, 32);
      p += __shfl_xor(p, 2, 32);
      p += __shfl_xor(p, 4, 32);
      p += __shfl_xor(p, 8, 32);
      if (n == 0) {                                  // lanes 0 and 16 write
        const int e = e0 + v + 8 * half;             // M = v + 8*half
        if (e < nedges) out[e] = p + b4v;
      }
    }
    __builtin_amdgcn_wave_barrier();                 // before next-tile gather
  }
}

extern "C" void kernel_launch(void* const* d_in, const int* in_sizes, int n_in,
                              void* d_out, int out_size, void* d_ws, size_t ws_size,
                              hipStream_t stream) {
  const float* feats = (const float*)d_in[0];
  const int*   eidx  = (const int*)d_in[1];
  const float* W1 = (const float*)d_in[2];
  const float* b1 = (const float*)d_in[3];
  const float* W2 = (const float*)d_in[4];
  const float* b2 = (const float*)d_in[5];
  const float* W3 = (const float*)d_in[6];
  const float* b3 = (const float*)d_in[7];
  const float* W4 = (const float*)d_in[8];
  const float* b4 = (const float*)d_in[9];
  float* out = (float*)d_out;

  const int nedges = in_sizes[1] / 2;                       // [2,E]
  const int ntiles = (nedges + TILE_M - 1) / TILE_M;        // 200,000 for E=3.2M
  const int nwaves = (ntiles + TILES_PER_WAVE - 1) / TILES_PER_WAVE;
  const int nblocks = (nwaves + WAVES_PER_BLOCK - 1) / WAVES_PER_BLOCK;

  edge_mlp_wmma<<<nblocks, WAVES_PER_BLOCK * 32, 0, stream>>>(
      feats, eidx, W1, b1, W2, b2, W3, b3, W4, b4, out, nedges, ntiles);
}